// WarpForward_10239202034200
// MI455X (gfx1250) — compile-verified
//
#include <hip/hip_runtime.h>

// Bilinear warp: out[p,f,i,j] = bilinear sample of x[p] at (i + u[...,1], j + u[...,0]),
// zero outside the image.
//
// Shapes: x (8,320,320) f32, u (8,12,320,320,2) f32, out (8,12,320,320) f32.
//
// Memory-bound streaming-gather kernel tuned for MI455X (gfx1250):
//  - 4 output pixels per thread -> 2x global_load_b128 NT (flow) + 1x global_store_b128 NT
//  - NT temporal hints on streaming operands keep the 3.3 MB image hot in WGP$/L2
//    for the 16 cached b32 gathers per thread
//  - geometry encoded in grid/block shape: zero per-thread integer divisions,
//    image index n is wave-uniform -> /12 and image base address are SALU/scalar
//  - no bounds checks: grid covers the domain exactly, EXEC stays full

typedef float v4f __attribute__((ext_vector_type(4)));

#define P_ 8
#define F_ 12
#define M_ 320
#define N_ 320

__device__ __forceinline__ float sample_tex(const float* __restrict__ xb, int yi, int xi) {
    bool valid = ((unsigned)xi < (unsigned)N_) & ((unsigned)yi < (unsigned)M_);
    int yc = min(max(yi, 0), M_ - 1);
    int xc = min(max(xi, 0), N_ - 1);
    float v = xb[yc * N_ + xc];   // regular-temporal load: image stays cached
    return valid ? v : 0.0f;
}

// block = (80, 4): tx = quad within row, ty = row within 4-row band (320 thr = 10 waves)
// grid  = (80, 96): bx = 4-row band, by = n = p*F + f  (wave-uniform)
__global__ __launch_bounds__(320) void warp_bilinear_kernel(
    const float* __restrict__ x,
    const float* __restrict__ u,
    float* __restrict__ out) {

    const int j0 = threadIdx.x << 2;                       // 0..316, step 4
    const int i  = (blockIdx.x << 2) + threadIdx.y;        // 0..319
    const int n  = blockIdx.y;                             // uniform: p*F + f
    const int p  = n / F_;                                 // uniform -> SALU

    const float* __restrict__ xb = x + (size_t)p * (M_ * N_);  // scalar base

    const size_t pixbase = ((size_t)n * M_ + i) * N_ + j0;

    // Flow for 4 pixels = 8 contiguous floats = 2 x b128, streamed (NT).
    const v4f* up = (const v4f*)(u + pixbase * 2);
    v4f ua = __builtin_nontemporal_load(up);
    v4f ub = __builtin_nontemporal_load(up + 1);

    float dxs[4] = {ua.x, ua.z, ub.x, ub.z};
    float dys[4] = {ua.y, ua.w, ub.y, ub.w};

    v4f res;
#pragma unroll
    for (int k = 0; k < 4; ++k) {
        float sx = (float)(j0 + k) + dxs[k];
        float sy = (float)i        + dys[k];
        float fx0 = floorf(sx);
        float fy0 = floorf(sy);
        float wx = sx - fx0;
        float wy = sy - fy0;
        int x0 = (int)fx0;
        int y0 = (int)fy0;

        float v00 = sample_tex(xb, y0,     x0    );
        float v01 = sample_tex(xb, y0,     x0 + 1);
        float v10 = sample_tex(xb, y0 + 1, x0    );
        float v11 = sample_tex(xb, y0 + 1, x0 + 1);

        // Same weight expansion as the reference for minimal numeric divergence.
        res[k] = v00 * (1.0f - wx) * (1.0f - wy)
               + v01 * wx          * (1.0f - wy)
               + v10 * (1.0f - wx) * wy
               + v11 * wx          * wy;
    }

    // Output for 4 pixels = b128 store, streamed (NT).
    __builtin_nontemporal_store(res, (v4f*)(out + pixbase));
}

extern "C" void kernel_launch(void* const* d_in, const int* in_sizes, int n_in,
                              void* d_out, int out_size, void* d_ws, size_t ws_size,
                              hipStream_t stream) {
    (void)in_sizes; (void)n_in; (void)out_size; (void)d_ws; (void)ws_size;

    const float* x   = (const float*)d_in[0];   // (8,320,320) f32
    const float* u   = (const float*)d_in[1];   // (8,12,320,320,2) f32
    float*       out = (float*)d_out;           // (8,12,320,320) f32

    dim3 block(N_ / 4, 4);       // (80,4) = 320 threads = 10 wave32s
    dim3 grid(M_ / 4, P_ * F_);  // (80,96): exact coverage, no tail

    warp_bilinear_kernel<<<grid, block, 0, stream>>>(x, u, out);
}